// UltrametricDistanceMatrix_33277406609659
// MI455X (gfx1250) — compile-verified
//
#include <hip/hip_runtime.h>
#include <math.h>

typedef __attribute__((ext_vector_type(2))) float v2f;
typedef __attribute__((ext_vector_type(8))) float v8f;

#define N      256
#define DIM    64
#define BATCH  4
#define EPS_U  1e-6f

// ---------------------------------------------------------------------------
// ws layout: [0 .. BATCH*N) floats   : row squared-norms
//            then BATCH uints        : ultrametricity violation counters
// ---------------------------------------------------------------------------

__global__ void row_norms_kernel(const float* __restrict__ X,
                                 float* __restrict__ norms) {
    int b = blockIdx.x;       // batch
    int r = threadIdx.x;      // row
    const float* row = X + ((size_t)b * N + r) * DIM;
    float s = 0.f;
#pragma unroll
    for (int d = 0; d < DIM; d += 4) {
        float4 v = *(const float4*)(row + d);
        s += v.x * v.x + v.y * v.y + v.z * v.z + v.w * v.w;
    }
    norms[b * N + r] = s;
}

// One wave32 per 16x16 output tile. Gram matrix via V_WMMA_F32_16X16X4_F32
// (fp32-exact matrix path on CDNA5), K=64 in 16 steps of 4.
//
// A (16x4, MxK) fragment per ISA: lane<16 holds row M=lane, K={k0,k0+1};
// lane>=16 holds row M=lane-16, K={k0+2,k0+3}.  B (4x16, KxN) is the mirror
// with N=lane&15.  Since B = X^T of the column block, both fragments load
// X[base + (lane&15)][k0 + 2*(lane>>4) + {0,1}] -> one 8B load each.
__global__ void euclid_wmma_kernel(const float* __restrict__ X,
                                   const float* __restrict__ norms,
                                   float* __restrict__ E) {
    const int lane  = threadIdx.x;        // 0..31, full wave (EXEC all 1s)
    const int jbase = blockIdx.x * 16;    // tile col block
    const int ibase = blockIdx.y * 16;    // tile row block
    const int b     = blockIdx.z;

    const int mrow = lane & 15;
    const int koff = (lane >> 4) * 2;

    const float* Xb = X + (size_t)b * N * DIM;
    const float* aP = Xb + (size_t)(ibase + mrow) * DIM + koff;
    const float* bP = Xb + (size_t)(jbase + mrow) * DIM + koff;

    v8f acc = {};
#pragma unroll
    for (int k0 = 0; k0 < DIM; k0 += 4) {
        v2f a, bb;
        a.x  = aP[k0 + 0];
        a.y  = aP[k0 + 1];
        bb.x = bP[k0 + 0];
        bb.y = bP[k0 + 1];
        // (neg_a, A, neg_b, B, c_mod, C, reuse_a, reuse_b)
        acc = __builtin_amdgcn_wmma_f32_16x16x4_f32(
            false, a, false, bb, (short)0, acc, false, false);
    }

    // C/D layout: VGPR v, lane l -> (M = v + 8*(l>>4), N = l&15)
    const int half = lane >> 4;
    const int col  = lane & 15;
    const float nJ = norms[b * N + jbase + col];
#pragma unroll
    for (int v = 0; v < 8; ++v) {
        int   m  = v + half * 8;
        float nI = norms[b * N + ibase + m];
        float sq = nI + nJ - 2.0f * acc[v];
        float d  = (sq > 0.f) ? sqrtf(sq) : 0.f;
        E[((size_t)b * N + ibase + m) * N + jbase + col] = d;
    }
}

__global__ void padic_kernel(float* __restrict__ Pout) {
    int i = blockIdx.x;
    int j = threadIdx.x;
    int d = (i > j) ? (i - j) : (j - i);
    float val = 0.f;
    if (d > 0) {
        int best = 0;
        int pk = 1;
#pragma unroll
        for (int k = 1; k <= 9; ++k) {
            pk *= 3;
            if (d % pk == 0) best = k;
        }
        val = 1.0f;
        for (int q = 0; q < best; ++q) val *= (1.0f / 3.0f);
    }
    Pout[i * N + j] = val;
}

__global__ void init_counters_kernel(unsigned int* __restrict__ viol) {
    if (threadIdx.x < BATCH) viol[threadIdx.x] = 0u;
}

// block = (b, i); thread j; loop k > j.  Rows i and j of D stream from L2.
__global__ void ultra_count_kernel(const float* __restrict__ E,
                                   unsigned int* __restrict__ viol) {
    const int i = blockIdx.x;
    const int b = blockIdx.y;
    const int j = threadIdx.x;
    const float* D = E + (size_t)b * N * N;

    int cnt = 0;
    if (j > i) {
        const float  dij  = D[(size_t)i * N + j];
        const float* rowJ = D + (size_t)j * N;
        const float* rowI = D + (size_t)i * N;
        for (int k = j + 1; k < N; ++k) {
            float djk = rowJ[k];
            float dik = rowI[k];
            float mx  = fmaxf(dij, fmaxf(djk, dik));
            float mn  = fminf(dij, fminf(djk, dik));
            float mid = dij + djk + dik - mx - mn;
            cnt += (mx > mid + EPS_U) ? 1 : 0;
        }
    }

    __shared__ int sdata[N];
    sdata[threadIdx.x] = cnt;
    __syncthreads();
#pragma unroll
    for (int s = N / 2; s > 0; s >>= 1) {
        if (threadIdx.x < s) sdata[threadIdx.x] += sdata[threadIdx.x + s];
        __syncthreads();
    }
    if (threadIdx.x == 0 && sdata[0] != 0)
        atomicAdd(&viol[b], (unsigned int)sdata[0]);
}

__global__ void finalize_kernel(const unsigned int* __restrict__ viol,
                                float* __restrict__ U) {
    int t = threadIdx.x;
    if (t < BATCH) {
        const float total = (float)((N * (N - 1) * (N - 2)) / 6);
        U[t] = 1.0f - (float)viol[t] / total;
    }
}

extern "C" void kernel_launch(void* const* d_in, const int* in_sizes, int n_in,
                              void* d_out, int out_size, void* d_ws, size_t ws_size,
                              hipStream_t stream) {
    const float* states = (const float*)d_in[0];   // (4,256,64) fp32

    float* out    = (float*)d_out;
    float* euclid = out;                        // 4*256*256
    float* padic  = euclid + (size_t)BATCH * N * N;  // 256*256
    float* ultra  = padic + (size_t)N * N;           // 4

    float*        norms = (float*)d_ws;                             // 1024 floats
    unsigned int* viol  = (unsigned int*)((char*)d_ws +
                                          (size_t)BATCH * N * sizeof(float));

    // 1) row squared-norms
    row_norms_kernel<<<dim3(BATCH), dim3(N), 0, stream>>>(states, norms);

    // 2) Gram-based pairwise Euclidean via fp32 WMMA (one wave per 16x16 tile)
    euclid_wmma_kernel<<<dim3(N / 16, N / 16, BATCH), dim3(32), 0, stream>>>(
        states, norms, euclid);

    // 3) p-adic distance matrix (batch-independent)
    padic_kernel<<<dim3(N), dim3(N), 0, stream>>>(padic);

    // 4) ultrametricity: zero counters, count violations, finalize
    init_counters_kernel<<<1, 32, 0, stream>>>(viol);
    ultra_count_kernel<<<dim3(N, BATCH), dim3(N), 0, stream>>>(euclid, viol);
    finalize_kernel<<<1, 32, 0, stream>>>(viol, ultra);
}